// ConvNormAct_38697655337417
// MI455X (gfx1250) — compile-verified
//
#include <hip/hip_runtime.h>
#include <math.h>

// ---------------------------------------------------------------------------
// ConvNormAct (quantum-pixel elementwise + BatchNorm2d + ReLU + residual)
// x: (B=16, C=64, H=128, W=128) f32.  Bandwidth-bound; 3-pass pipeline that
// exploits MI455X's 192MB L2 (64MiB input stays resident between passes).
// Wave reduction uses V_WMMA_F32_16X16X4_F32 (CDNA5 wave32 WMMA path).
// ---------------------------------------------------------------------------

typedef __attribute__((ext_vector_type(2))) float v2f;
typedef __attribute__((ext_vector_type(8))) float v8f;

#define COS_RY     0.99500416527802576721f   // cos(0.1)
#define CHUNK      16384                     // H*W floats, contiguous per (b,c)
#define NCHUNKS    1024                      // B*C
#define NCHAN      64
#define NBATCH     16

__device__ __forceinline__ float qpix(float x) {
    float z = COS_RY * __cosf(x);
    return z * (1.0f + z * (1.0f + z * (1.0f + z)));   // z + z^2 + z^3 + z^4
}

// Sum `s` across all 32 lanes of a wave via one WMMA.
// A = (s, 0) per lane: row m of the 16x4 A holds lane m's and lane (m+16)'s
// two slots -> with B = ones(4x16), D[m,n] = s_m + s_{m+16} for every n.
// Lane L holds D[r + 8*(L>=16), L%16] in c[r] (documented C/D layout), so
// sum of the 8 regs + cross-half add (shfl_xor 16) = full 32-lane sum,
// replicated to every lane.  Requires EXEC all-ones (no divergence here).
__device__ __forceinline__ float wave_sum_wmma(float s) {
    v2f a; a.x = s;    a.y = 0.0f;
    v2f b; b.x = 1.0f; b.y = 1.0f;
    v8f c = {};
    c = __builtin_amdgcn_wmma_f32_16x16x4_f32(false, a, false, b,
                                              (short)0, c, false, false);
    float local = ((c[0] + c[1]) + (c[2] + c[3])) +
                  ((c[4] + c[5]) + (c[6] + c[7]));
    return local + __shfl_xor(local, 16, 32);
}

// Pass 1: per-(b,c) chunk partial sum / sumsq of q.  One block per chunk.
__global__ __launch_bounds__(256)
void qbn_reduce_kernel(const float* __restrict__ x,
                       float* __restrict__ psum,
                       float* __restrict__ psqs) {
    const int chunk = blockIdx.x;                       // b*C + c
    const float4* xc = (const float4*)(x + (size_t)chunk * CHUNK);
    const int tid = threadIdx.x;

    float s = 0.0f, t = 0.0f;
    #pragma unroll
    for (int i = 0; i < 16; ++i) {                      // 256 thr * 16 * 4 = 16384
        float4 v = xc[tid + i * 256];
        float q0 = qpix(v.x), q1 = qpix(v.y), q2 = qpix(v.z), q3 = qpix(v.w);
        s += (q0 + q1) + (q2 + q3);
        t += (q0 * q0 + q1 * q1) + (q2 * q2 + q3 * q3);
    }

    // Wave32 reduction on the matrix pipe (all 256 lanes active -> EXEC ok).
    float S = wave_sum_wmma(s);
    float T = wave_sum_wmma(t);

    __shared__ float sS[8], sT[8];
    const int lane = tid & 31, wave = tid >> 5;
    if (lane == 0) { sS[wave] = S; sT[wave] = T; }
    __syncthreads();
    if (tid == 0) {
        float a = 0.0f;
        #pragma unroll
        for (int w = 0; w < 8; ++w) a += sS[w];
        psum[chunk] = a;
    }
    if (tid == 1) {
        float a = 0.0f;
        #pragma unroll
        for (int w = 0; w < 8; ++w) a += sT[w];
        psqs[chunk] = a;
    }
}

// Pass 2: fold BN into per-channel affine y = q*a + b.  Tiny; f64 accumulate.
__global__ __launch_bounds__(64)
void qbn_stats_kernel(const float* __restrict__ psum,
                      const float* __restrict__ psqs,
                      const float* __restrict__ gamma,
                      const float* __restrict__ beta,
                      float* __restrict__ ab) {
    const int c = threadIdx.x;                          // 0..63
    double S = 0.0, T = 0.0;
    #pragma unroll
    for (int bidx = 0; bidx < NBATCH; ++bidx) {
        S += (double)psum[bidx * NCHAN + c];
        T += (double)psqs[bidx * NCHAN + c];
    }
    const double N    = (double)NBATCH * (double)CHUNK; // 262144
    const double mean = S / N;
    double var = T / N - mean * mean;
    if (var < 0.0) var = 0.0;
    const float rinv = (float)(1.0 / sqrt(var + 1e-6));
    const float a = gamma[c] * rinv;
    const float b = beta[c] - (float)mean * a;
    ab[c]         = a;
    ab[NCHAN + c] = b;
}

// Pass 3: y = relu(q*a_c + b_c) + x, float4 vectorized.  x re-read from L2.
__global__ __launch_bounds__(256)
void qbn_apply_kernel(const float* __restrict__ x,
                      const float* __restrict__ ab,
                      float* __restrict__ out) {
    const int idx4 = blockIdx.x * 256 + threadIdx.x;    // float4 index
    const int c = (idx4 >> 12) & (NCHAN - 1);           // CHUNK/4 = 4096 f4/chunk
    const float a = ab[c];
    const float b = ab[NCHAN + c];
    const float4 v = ((const float4*)x)[idx4];
    float4 r;
    r.x = fmaxf(fmaf(qpix(v.x), a, b), 0.0f) + v.x;
    r.y = fmaxf(fmaf(qpix(v.y), a, b), 0.0f) + v.y;
    r.z = fmaxf(fmaf(qpix(v.z), a, b), 0.0f) + v.z;
    r.w = fmaxf(fmaf(qpix(v.w), a, b), 0.0f) + v.w;
    ((float4*)out)[idx4] = r;
}

extern "C" void kernel_launch(void* const* d_in, const int* in_sizes, int n_in,
                              void* d_out, int out_size, void* d_ws, size_t ws_size,
                              hipStream_t stream) {
    const float* x     = (const float*)d_in[0];   // 16*64*128*128
    const float* gamma = (const float*)d_in[1];   // 64
    const float* beta  = (const float*)d_in[2];   // 64
    float* out = (float*)d_out;

    // Workspace layout (floats): psum[1024] | psqs[1024] | a[64] | b[64]
    float* ws   = (float*)d_ws;
    float* psum = ws;
    float* psqs = ws + NCHUNKS;
    float* ab   = ws + 2 * NCHUNKS;

    qbn_reduce_kernel<<<NCHUNKS, 256, 0, stream>>>(x, psum, psqs);
    qbn_stats_kernel<<<1, 64, 0, stream>>>(psum, psqs, gamma, beta, ab);
    qbn_apply_kernel<<<(16777216 / 4) / 256, 256, 0, stream>>>(x, ab, out);
}